// AttnDecoderRNN_63763084476991
// MI455X (gfx1250) — compile-verified
//
#include <hip/hip_runtime.h>
#include <hip/hip_bf16.h>

// AttnDecoderRNN forward for MI455X (gfx1250), fp32 WMMA path.
// N=1024, L=128, H=512, O=32000.

typedef __attribute__((ext_vector_type(2))) float v2f;
typedef __attribute__((ext_vector_type(8))) float v8f;

#define N_BATCH 1024
#define L_LEN   128
#define H_DIM   512
#define O_DIM   32000

#define WMMA_F32(Aop, Bop, Cop) \
  __builtin_amdgcn_wmma_f32_16x16x4_f32(false, (Aop), false, (Bop), (short)0, (Cop), false, false)

#define FOR8(M) M(0) M(1) M(2) M(3) M(4) M(5) M(6) M(7)

// ---------------------------------------------------------------------------
// Kernel 1: embedded = emb[ids]; cat_eh = [embedded | hidden]; comb_in[:, :H] = embedded
// ---------------------------------------------------------------------------
__global__ void embed_cat_kernel(const int* __restrict__ ids,
                                 const float* __restrict__ emb,
                                 const float* __restrict__ hidden,
                                 float* __restrict__ cat_eh,
                                 float* __restrict__ comb_in) {
  int n = blockIdx.x;
  int t = threadIdx.x;  // 0..127
  const float4* erow = (const float4*)(emb + (size_t)ids[n] * H_DIM);
  const float4* hrow = (const float4*)(hidden + (size_t)n * H_DIM);
  float4 e = erow[t];
  float4 h = hrow[t];
  float4* cat = (float4*)(cat_eh + (size_t)n * (2 * H_DIM));
  cat[t] = e;
  cat[(H_DIM / 4) + t] = h;
  float4* cmb = (float4*)(comb_in + (size_t)n * (2 * H_DIM));
  cmb[t] = e;
}

// ---------------------------------------------------------------------------
// Generic fp32 WMMA GEMM: C[M,N] = act(A[M,K] @ B[N,K]^T + bias[N])
// Wave tile: 16(M) x 128(N), 8 accumulators, K-step 4 (V_WMMA_F32_16X16X4_F32).
//
// Software pipeline: ping-pong fragment buffers (bX/bY), no rotation copies.
// Each iteration covers k and k+4:
//   load Y(k+4)  [imm +16B] ; 8 WMMAs on X (loaded a full stage earlier)
//   load X(k+8)  [imm +32B] ; 8 WMMAs on Y ; bump 9 row pointers by 32B
// B-row pointers are 8 *named* scalar variables (not an array) so clang's
// address-space inference keeps them global -> global_load_b64, not
// flat_load_b64 (FLAT would also tie up DScnt + the LDS path).
// Requires M%16==0, N%128==0, K%8==0; grid exact so EXEC is all-ones.
//
// fp32 A (16x4) lane layout (ISA 7.12.2): lane L holds row M=L%16,
//   K = 2*(L/16)+{0,1} -> per-lane float2 at A[row*K + k + 2*(L/16)]
// fp32 B (4x16): mirrored. fp32 C/D 16x16: VGPR v, lane L ->
//   C[M = v + 8*(L/16)][N = L%16]
// ---------------------------------------------------------------------------
__global__ void gemm_nt_f32_wmma(const float* __restrict__ A,
                                 const float* __restrict__ B,
                                 const float* __restrict__ bias,
                                 float* __restrict__ C,
                                 int M, int N, int K, int relu) {
  int wid  = (blockIdx.x * blockDim.x + threadIdx.x) >> 5;
  int lane = threadIdx.x & 31;
  int n128 = N >> 7;
  int mt   = wid / n128;
  int nt   = wid - mt * n128;
  int hh   = lane >> 4;   // 0 or 1
  int l16  = lane & 15;
  int koff = hh * 2;

  const float* ap  = A + (size_t)(mt * 16 + l16) * K + koff;
  const float* bp0base = B + (size_t)(nt * 128 + l16) * K + koff;

#define B_PTR_DECL(i) const float* bq##i = bp0base + (size_t)(16 * (i)) * K;
  FOR8(B_PTR_DECL)
#undef B_PTR_DECL

  v8f acc[8];
#pragma unroll
  for (int j = 0; j < 8; ++j) acc[j] = (v8f){};

  v2f aX, aY, bX[8], bY[8];

  // ---- prologue: stage X at k=0
  aX = *(const v2f*)(ap);
#define LOAD_X0(i) bX[i] = *(const v2f*)(bq##i);
  FOR8(LOAD_X0)
#undef LOAD_X0

  // ---- steady state: each trip covers k, k+4; prefetches k+8
  for (int k = 0; k < K - 8; k += 8) {
    aY = *(const v2f*)(ap + 4);
#define LOAD_Y(i) bY[i] = *(const v2f*)(bq##i + 4);
    FOR8(LOAD_Y)
#undef LOAD_Y

#pragma unroll
    for (int j = 0; j < 8; ++j) acc[j] = WMMA_F32(aX, bX[j], acc[j]);

    aX = *(const v2f*)(ap + 8);
#define LOAD_XN(i) bX[i] = *(const v2f*)(bq##i + 8);
    FOR8(LOAD_XN)
#undef LOAD_XN

#pragma unroll
    for (int j = 0; j < 8; ++j) acc[j] = WMMA_F32(aY, bY[j], acc[j]);

    ap += 8;
#define BUMP(i) bq##i += 8;
    FOR8(BUMP)
#undef BUMP
  }

  // ---- tail: k = K-8 (X staged), plus k = K-4
  aY = *(const v2f*)(ap + 4);
#define LOAD_YT(i) bY[i] = *(const v2f*)(bq##i + 4);
  FOR8(LOAD_YT)
#undef LOAD_YT
#pragma unroll
  for (int j = 0; j < 8; ++j) acc[j] = WMMA_F32(aX, bX[j], acc[j]);
#pragma unroll
  for (int j = 0; j < 8; ++j) acc[j] = WMMA_F32(aY, bY[j], acc[j]);

  // ---- store with bias (+ optional relu)
#pragma unroll
  for (int j = 0; j < 8; ++j) {
    int col  = nt * 128 + j * 16 + l16;
    float bv = bias[col];
#pragma unroll
    for (int v = 0; v < 8; ++v) {
      float cv = acc[j][v] + bv;
      if (relu) cv = fmaxf(cv, 0.0f);
      C[(size_t)(mt * 16 + v + hh * 8) * N + col] = cv;
    }
  }
}

// ---------------------------------------------------------------------------
// Kernel 3: row softmax over L=128. One 128-thread block per row.
// ---------------------------------------------------------------------------
__global__ void softmax_l_kernel(const float* __restrict__ logits,
                                 float* __restrict__ out) {
  __shared__ float red[L_LEN];
  int n = blockIdx.x, t = threadIdx.x;
  float x = logits[(size_t)n * L_LEN + t];
  red[t] = x;
  __syncthreads();
  for (int s = L_LEN / 2; s > 0; s >>= 1) {
    if (t < s) red[t] = fmaxf(red[t], red[t + s]);
    __syncthreads();
  }
  float m = red[0];
  __syncthreads();
  float e = expf(x - m);
  red[t] = e;
  __syncthreads();
  for (int s = L_LEN / 2; s > 0; s >>= 1) {
    if (t < s) red[t] += red[t + s];
    __syncthreads();
  }
  out[(size_t)n * L_LEN + t] = e / red[0];
}

// ---------------------------------------------------------------------------
// Kernel 4: attn_applied[n,h] = sum_l w[n,l] * enc[n,l,h] * mask[n,l,h]
// written into comb_in[n, H + h]. Streams 512 MB with float4 loads.
// ---------------------------------------------------------------------------
__global__ void attn_apply_kernel(const float* __restrict__ attn_w,
                                  const float* __restrict__ enc,
                                  const float* __restrict__ mask,
                                  float* __restrict__ comb_in) {
  __shared__ float w[L_LEN];
  int n = blockIdx.x, t = threadIdx.x;  // 128 threads
  w[t] = attn_w[(size_t)n * L_LEN + t];
  __syncthreads();
  float4 acc = make_float4(0.f, 0.f, 0.f, 0.f);
  const float4* E  = (const float4*)(enc  + (size_t)n * L_LEN * H_DIM);
  const float4* Mk = (const float4*)(mask + (size_t)n * L_LEN * H_DIM);
  for (int l = 0; l < L_LEN; ++l) {
    float4 e = E[l * (H_DIM / 4) + t];
    float4 m = Mk[l * (H_DIM / 4) + t];
    float wl = w[l];
    acc.x += wl * e.x * m.x;
    acc.y += wl * e.y * m.y;
    acc.z += wl * e.z * m.z;
    acc.w += wl * e.w * m.w;
  }
  float4* dst = (float4*)(comb_in + (size_t)n * (2 * H_DIM) + H_DIM);
  dst[t] = acc;
}

// ---------------------------------------------------------------------------
// Kernel 7: GRU pointwise gates (PyTorch order r,z,n). Biases already folded
// into gx/gh by the GEMM epilogues.
// ---------------------------------------------------------------------------
__global__ void gru_kernel(const float* __restrict__ gx,
                           const float* __restrict__ gh,
                           const float* __restrict__ hidden,
                           float* __restrict__ h_new) {
  int idx = blockIdx.x * blockDim.x + threadIdx.x;  // < N_BATCH*H_DIM
  int n = idx >> 9;       // / H_DIM
  int j = idx & (H_DIM - 1);
  const float* gxr = gx + (size_t)n * 3 * H_DIM;
  const float* ghr = gh + (size_t)n * 3 * H_DIM;
  float r  = 1.0f / (1.0f + expf(-(gxr[j] + ghr[j])));
  float z  = 1.0f / (1.0f + expf(-(gxr[H_DIM + j] + ghr[H_DIM + j])));
  float nn = tanhf(gxr[2 * H_DIM + j] + r * ghr[2 * H_DIM + j]);
  float h  = hidden[idx];
  h_new[idx] = (1.0f - z) * nn + z * h;
}

// ---------------------------------------------------------------------------
// Kernel 9: in-place log_softmax over O=32000. One 256-thread block per row.
// Row is 128 KB -> stays L2-resident across the three passes.
// ---------------------------------------------------------------------------
__global__ void log_softmax_kernel(float* __restrict__ out) {
  __shared__ float red[256];
  int n = blockIdx.x, t = threadIdx.x;
  float* row = out + (size_t)n * O_DIM;
  float m = -3.402823466e38f;
  for (int i = t; i < O_DIM; i += 256) m = fmaxf(m, row[i]);
  red[t] = m;
  __syncthreads();
  for (int s = 128; s > 0; s >>= 1) {
    if (t < s) red[t] = fmaxf(red[t], red[t + s]);
    __syncthreads();
  }
  m = red[0];
  __syncthreads();
  float sum = 0.0f;
  for (int i = t; i < O_DIM; i += 256) sum += expf(row[i] - m);
  red[t] = sum;
  __syncthreads();
  for (int s = 128; s > 0; s >>= 1) {
    if (t < s) red[t] += red[t + s];
    __syncthreads();
  }
  float lse = m + logf(red[0]);
  for (int i = t; i < O_DIM; i += 256) row[i] = row[i] - lse;
}

// ---------------------------------------------------------------------------
extern "C" void kernel_launch(void* const* d_in, const int* in_sizes, int n_in,
                              void* d_out, int out_size, void* d_ws, size_t ws_size,
                              hipStream_t stream) {
  const int*   ids    = (const int*)d_in[0];
  const float* hidden = (const float*)d_in[1];
  const float* enc    = (const float*)d_in[2];
  const float* mask   = (const float*)d_in[3];
  const float* emb    = (const float*)d_in[4];
  const float* attn_W = (const float*)d_in[5];
  const float* attn_b = (const float*)d_in[6];
  const float* comb_W = (const float*)d_in[7];
  const float* comb_b = (const float*)d_in[8];
  const float* W_ih   = (const float*)d_in[9];
  const float* W_hh   = (const float*)d_in[10];
  const float* b_ih   = (const float*)d_in[11];
  const float* b_hh   = (const float*)d_in[12];
  const float* out_W  = (const float*)d_in[13];
  const float* out_b  = (const float*)d_in[14];

  // Output layout (tuple concat): [output N*O][h_new N*H][attn_weights N*L]
  float* out_logits = (float*)d_out;
  float* out_hnew   = out_logits + (size_t)N_BATCH * O_DIM;
  float* out_attnw  = out_hnew   + (size_t)N_BATCH * H_DIM;

  // Workspace carving (~24 MB of floats)
  float* ws       = (float*)d_ws;
  float* cat_eh   = ws;                                      // N*2H
  float* comb_in  = cat_eh  + (size_t)N_BATCH * 2 * H_DIM;   // N*2H
  float* logits_l = comb_in + (size_t)N_BATCH * 2 * H_DIM;   // N*L
  float* xbuf     = logits_l + (size_t)N_BATCH * L_LEN;      // N*H
  float* gx       = xbuf    + (size_t)N_BATCH * H_DIM;       // N*3H
  float* gh       = gx      + (size_t)N_BATCH * 3 * H_DIM;   // N*3H

  const int MT = N_BATCH / 16;  // 64 M-tiles; waves per GEMM = MT * (N/128)

  // 1) gather + concat
  embed_cat_kernel<<<N_BATCH, 128, 0, stream>>>(ids, emb, hidden, cat_eh, comb_in);

  // 2) attn logits = cat_eh @ attn_W^T + attn_b   (M=1024, N=128, K=1024)
  gemm_nt_f32_wmma<<<(MT * (L_LEN / 128)) / 8, 256, 0, stream>>>(
      cat_eh, attn_W, attn_b, logits_l, N_BATCH, L_LEN, 2 * H_DIM, 0);

  // 3) softmax over L -> attn_weights output
  softmax_l_kernel<<<N_BATCH, L_LEN, 0, stream>>>(logits_l, out_attnw);

  // 4) attn_applied (with mask fused) -> comb_in[:, H:2H]
  attn_apply_kernel<<<N_BATCH, 128, 0, stream>>>(out_attnw, enc, mask, comb_in);

  // 5) x = relu(comb_in @ comb_W^T + comb_b)      (M=1024, N=512, K=1024)
  gemm_nt_f32_wmma<<<(MT * (H_DIM / 128)) / 8, 256, 0, stream>>>(
      comb_in, comb_W, comb_b, xbuf, N_BATCH, H_DIM, 2 * H_DIM, 1);

  // 6) GRU input/hidden projections (M=1024, N=1536, K=512)
  gemm_nt_f32_wmma<<<(MT * (3 * H_DIM / 128)) / 8, 256, 0, stream>>>(
      xbuf, W_ih, b_ih, gx, N_BATCH, 3 * H_DIM, H_DIM, 0);
  gemm_nt_f32_wmma<<<(MT * (3 * H_DIM / 128)) / 8, 256, 0, stream>>>(
      hidden, W_hh, b_hh, gh, N_BATCH, 3 * H_DIM, H_DIM, 0);

  // 7) GRU gates -> h_new output
  gru_kernel<<<(N_BATCH * H_DIM) / 256, 256, 0, stream>>>(gx, gh, hidden, out_hnew);

  // 8) logits = h_new @ out_W^T + out_b           (M=1024, N=32000, K=512)
  gemm_nt_f32_wmma<<<(MT * (O_DIM / 128)) / 8, 256, 0, stream>>>(
      out_hnew, out_W, out_b, out_logits, N_BATCH, O_DIM, H_DIM, 0);

  // 9) in-place log_softmax over O
  log_softmax_kernel<<<N_BATCH, 256, 0, stream>>>(out_logits);
}